// TwoStateExplorer_64561948393589
// MI455X (gfx1250) — compile-verified
//
#include <hip/hip_runtime.h>
#include <hip/hip_bf16.h>
#include <stdint.h>

#define BB   512
#define KKE  256
#define SPD  0.005f
#define DTT  0.1f

typedef __attribute__((ext_vector_type(2))) float v2f;
typedef __attribute__((ext_vector_type(8))) float v8f;
typedef __attribute__((ext_vector_type(4))) unsigned int u32x4;
typedef __attribute__((ext_vector_type(8))) int i32x8;
typedef __attribute__((ext_vector_type(4))) int i32x4;

// per-particle staging block in d_ws (floats): 2048 floats = 8 KB
#define L_TUM 0
#define L_SEG 256
#define L_E0  512      // e0x[256] e0y[256] e0z[256]
#define L_PX  1280     // px[256] py[256] pz[256]
#define BLK   2048

// ---------------------------------------------------------------- helpers
__device__ __forceinline__ void norm3(float& x, float& y, float& z) {
    float r = rsqrtf(x * x + y * y + z * z);
    x *= r; y *= r; z *= r;
}
__device__ __forceinline__ void rodrigues(float ux, float uy, float uz, float ang,
                                          float& vx, float& vy, float& vz) {
    float c = __cosf(ang), s = __sinf(ang);
    float cx = uy * vz - uz * vy;
    float cy = uz * vx - ux * vz;
    float cz = ux * vy - uy * vx;
    float t  = (1.0f - c) * (ux * vx + uy * vy + uz * vz);
    vx = c * vx + s * cx + t * ux;
    vy = c * vy + s * cy + t * uy;
    vz = c * vz + s * cz + t * uz;
}

// ------------------------------------------------ kernel 1: frame rotations
__global__ void kernelFrames(const float* __restrict__ e0r, const float* __restrict__ e1r,
                             const float* __restrict__ pa, const float* __restrict__ na,
                             float* __restrict__ ws) {
    int b = blockIdx.x * blockDim.x + threadIdx.x;
    if (b >= BB) return;
    float e0x = e0r[3 * b + 0], e0y = e0r[3 * b + 1], e0z = e0r[3 * b + 2];
    float e1x = e1r[3 * b + 0], e1y = e1r[3 * b + 1], e1z = e1r[3 * b + 2];
    norm3(e0x, e0y, e0z);
    {   // orth(e1, e0) then norm
        float d = e1x * e0x + e1y * e0y + e1z * e0z;
        e1x -= d * e0x; e1y -= d * e0y; e1z -= d * e0z;
        norm3(e1x, e1y, e1z);
    }
    float e2x = e0y * e1z - e0z * e1y;
    float e2y = e0z * e1x - e0x * e1z;
    float e2z = e0x * e1y - e0y * e1x;
    norm3(e2x, e2y, e2z);

    float* blk = ws + (size_t)b * BLK;
    for (int k = 0; k < KKE; ++k) {
        float p = pa[b * KKE + k];
        float q = na[b * KKE + k];
        // planar rotation about e2 (old e2 for both)
        rodrigues(e2x, e2y, e2z, p, e0x, e0y, e0z); norm3(e0x, e0y, e0z);
        rodrigues(e2x, e2y, e2z, p, e1x, e1y, e1z); norm3(e1x, e1y, e1z);
        { float d = e1x * e0x + e1y * e0y + e1z * e0z;
          e1x -= d * e0x; e1y -= d * e0y; e1z -= d * e0z; norm3(e1x, e1y, e1z); }
        e2x = e0y * e1z - e0z * e1y; e2y = e0z * e1x - e0x * e1z; e2z = e0x * e1y - e0y * e1x;
        norm3(e2x, e2y, e2z);
        // nonplanar rotation about e1
        rodrigues(e1x, e1y, e1z, q, e0x, e0y, e0z); norm3(e0x, e0y, e0z);
        { float d = e1x * e0x + e1y * e0y + e1z * e0z;
          e1x -= d * e0x; e1y -= d * e0y; e1z -= d * e0z; norm3(e1x, e1y, e1z); }
        e2x = e0y * e1z - e0z * e1y; e2y = e0z * e1x - e0x * e1z; e2z = e0x * e1y - e0y * e1x;
        norm3(e2x, e2y, e2z);

        blk[L_E0 +       k] = e0x;
        blk[L_E0 + 256 + k] = e0y;
        blk[L_E0 + 512 + k] = e0z;
    }
}

// ------------------------------------------------ kernel 2: WMMA cumsums
__device__ __forceinline__ v8f wmma4(v2f a, v2f b, v8f c) {
    return __builtin_amdgcn_wmma_f32_16x16x4_f32(false, a, false, b, (short)0, c,
                                                 false, false);
}

// Inclusive 256-element cumsum for one wave32.
// View v as V[r][c] = v[16r+c]; S = (Lstrict*V)*J + V*U  (all 16x16 f32 matmuls,
// each K=16 matmul = 4 chained V_WMMA_F32_16X16X4_F32).
__device__ __forceinline__ v8f wave_cumsum256(float* lds_v, float* lds_p, int lane) {
    const int n   = lane & 15;         // column / M index
    const int hi  = (lane >> 4) & 1;   // half-wave select
    const int kl0 = hi * 2;            // local K (A) / local row (B) for vgpr0

    v8f acc = {0.f, 0.f, 0.f, 0.f, 0.f, 0.f, 0.f, 0.f};
#pragma unroll
    for (int kk = 0; kk < 4; ++kk) {   // P = Lstrict x V
        v2f a, b;
        a.x = (n > 4 * kk + kl0)     ? 1.0f : 0.0f;
        a.y = (n > 4 * kk + kl0 + 1) ? 1.0f : 0.0f;
        b.x = lds_v[16 * (4 * kk + kl0)     + n];
        b.y = lds_v[16 * (4 * kk + kl0 + 1) + n];
        acc = wmma4(a, b, acc);
    }
#pragma unroll
    for (int g = 0; g < 8; ++g)        // spill P row-major to LDS
        lds_p[16 * (g + 8 * hi) + n] = acc[g];

    v8f s = {0.f, 0.f, 0.f, 0.f, 0.f, 0.f, 0.f, 0.f};
#pragma unroll
    for (int kk = 0; kk < 4; ++kk) {   // s += V x U  (U upper-tri ones)
        v2f a, b;
        a.x = lds_v[16 * n + 4 * kk + kl0];
        a.y = lds_v[16 * n + 4 * kk + kl0 + 1];
        b.x = (4 * kk + kl0     <= n) ? 1.0f : 0.0f;
        b.y = (4 * kk + kl0 + 1 <= n) ? 1.0f : 0.0f;
        s = wmma4(a, b, s);
    }
#pragma unroll
    for (int kk = 0; kk < 4; ++kk) {   // s += P x J  (J all-ones)
        v2f a, b;
        a.x = lds_p[16 * n + 4 * kk + kl0];
        a.y = lds_p[16 * n + 4 * kk + kl0 + 1];
        b.x = 1.0f; b.y = 1.0f;
        s = wmma4(a, b, s);
    }
    return s;                          // inclusive cumsum, C/D layout
}

__global__ void kernelW(const float* __restrict__ rd_in, float* __restrict__ ws,
                        float* __restrict__ out_tumble) {
    __shared__ float smem[8][512];
    const int wave = threadIdx.x >> 5;
    const int lane = threadIdx.x & 31;
    const int b    = blockIdx.x * 8 + wave;
    const int n    = lane & 15;
    const int hi   = (lane >> 4) & 1;

    float* lds_v = &smem[wave][0];
    float* lds_p = &smem[wave][256];
    float* blk   = ws + (size_t)b * BLK;
    const float* rd = rd_in + (size_t)b * KKE;

    // ---- array 0: run_durations -> tumble_ts (incl), seg_start (excl)
#pragma unroll
    for (int i = 0; i < 8; ++i) lds_v[lane + 32 * i] = rd[lane + 32 * i];
    v8f S = wave_cumsum256(lds_v, lds_p, lane);
#pragma unroll
    for (int g = 0; g < 8; ++g) {
        int nn  = 16 * (g + 8 * hi) + n;
        float s = S[g];
        blk[L_TUM + nn] = s;
        out_tumble[(size_t)b * KKE + nn] = s;
        blk[L_SEG + nn] = s - lds_v[nn];
    }
    // ---- arrays 1..3: SPEED*rd*e0c -> exclusive prefix displacement
    for (int c = 0; c < 3; ++c) {
        const float* e0c = blk + L_E0 + 256 * c;
#pragma unroll
        for (int i = 0; i < 8; ++i) {
            int idx = lane + 32 * i;
            lds_v[idx] = SPD * rd[idx] * e0c[idx];
        }
        v8f Sc = wave_cumsum256(lds_v, lds_p, lane);
#pragma unroll
        for (int g = 0; g < 8; ++g) {
            int nn = 16 * (g + 8 * hi) + n;
            blk[L_PX + 256 * c + nn] = Sc[g] - lds_v[nn];
        }
    }
}

// ------------------------------------------------ kernel 3: trajectory (TDM + search)
__global__ void kernelX(const float* __restrict__ ws, const float* __restrict__ x0,
                        float* __restrict__ out, int n_steps) {
    __shared__ float smem[BLK];
    const int b = blockIdx.y;

    if (threadIdx.x < 32) {            // wave 0 issues one 8 KB TDM tile load
        uint64_t ga  = (uint64_t)(uintptr_t)(ws + (size_t)b * BLK);
        uint32_t lds = (uint32_t)(uintptr_t)(&smem[0]);
        u32x4 g0 = {1u,                          // count=1, user descriptor
                    lds,                         // lds_addr
                    (uint32_t)ga,                // global_addr[31:0]
                    (uint32_t)((ga >> 32) & 0x01FFFFFFu) | 0x80000000u}; // addr[56:32]|type=2
        i32x8 g1 = {0x00020000,                  // wg_mask=0, data_size=4B
                    (int)(2048u << 16),          // tensor_dim0 = 2048
                    0x00010000,                  // tensor_dim1 = 1
                    (int)(2048u << 16),          // tile_dim0 = 2048
                    0,                           // tile_dim1 = tile_dim2 = 0 (1-D)
                    2048,                        // tensor_dim0_stride
                    0, 0};
        i32x4 g2 = {1, 1, 0, 0};                 // tensor_dim2=1, tensor_dim3=1
        i32x4 g3 = {0, 0x00010000, 0, 0};        // tensor_dim4=1
        i32x8 g4 = {0, 0, 0, 0, 0, 0, 0, 0};     // extra group (6-arg toolchain form)
        __builtin_amdgcn_tensor_load_to_lds(g0, g1, g2, g3, g4, 0);
        __builtin_amdgcn_s_wait_tensorcnt(0);
    }
    __syncthreads();

    const float* tum = smem + L_TUM;
    const float* seg = smem + L_SEG;
    const float* e0x = smem + L_E0;
    const float* e0y = smem + L_E0 + 256;
    const float* e0z = smem + L_E0 + 512;
    const float* px  = smem + L_PX;
    const float* py  = smem + L_PX + 256;
    const float* pz  = smem + L_PX + 512;

    const float X0 = x0[3 * b + 0], Y0 = x0[3 * b + 1], Z0 = x0[3 * b + 2];
    const size_t off2 = (size_t)n_steps + (size_t)BB * KKE;   // X region offset
    const int base = blockIdx.x * 1024 + threadIdx.x;

#pragma unroll
    for (int q = 0; q < 4; ++q) {
        int ti = base + q * 256;
        if (ti < n_steps) {
            float t = (float)ti * DTT;
            int j = 0;                           // searchsorted(side='right')
#pragma unroll
            for (int s = 128; s >= 1; s >>= 1)
                if (tum[j + s - 1] <= t) j += s;
            if (j > KKE - 1) j = KKE - 1;
            float dt = t - seg[j];
            float xx = X0 + px[j] + SPD * e0x[j] * dt;
            float yy = Y0 + py[j] + SPD * e0y[j] * dt;
            float zz = Z0 + pz[j] + SPD * e0z[j] * dt;
            size_t o = off2 + ((size_t)b * n_steps + ti) * 3;
            out[o + 0] = xx; out[o + 1] = yy; out[o + 2] = zz;
        }
    }
}

// ------------------------------------------------ kernel 4: ts + passthroughs
__global__ void kernelCopy(const float* __restrict__ rd, const float* __restrict__ pa,
                           const float* __restrict__ na, float* __restrict__ out,
                           int n_steps) {
    int i = blockIdx.x * blockDim.x + threadIdx.x;
    const int NK = BB * KKE;
    if (i < n_steps) out[i] = (float)i * DTT;
    if (i < NK) {
        size_t off3 = (size_t)n_steps + NK + (size_t)BB * n_steps * 3;
        out[off3 + i]            = rd[i];
        out[off3 + NK + i]       = pa[i];
        out[off3 + 2 * NK + i]   = na[i];
    }
}

// ------------------------------------------------ launcher
extern "C" void kernel_launch(void* const* d_in, const int* in_sizes, int n_in,
                              void* d_out, int out_size, void* d_ws, size_t ws_size,
                              hipStream_t stream) {
    const float* x0  = (const float*)d_in[1];
    const float* e0r = (const float*)d_in[2];
    const float* e1r = (const float*)d_in[3];
    const float* rd  = (const float*)d_in[4];
    const float* pa  = (const float*)d_in[5];
    const float* na  = (const float*)d_in[6];
    float* out = (float*)d_out;
    float* ws  = (float*)d_ws;

    // out = ts[n] | tumble[BK] | X[B,n,3] | rd | pa | na  =>  solve for n
    int n_steps = (int)(((long long)out_size - 4LL * BB * KKE) / (3LL * BB + 1LL));

    kernelFrames<<<2, 256, 0, stream>>>(e0r, e1r, pa, na, ws);
    kernelW<<<BB / 8, 256, 0, stream>>>(rd, ws, out + n_steps);
    dim3 gX((n_steps + 1023) / 1024, BB);
    kernelX<<<gX, 256, 0, stream>>>(ws, x0, out, n_steps);
    int nmax = (BB * KKE > n_steps) ? BB * KKE : n_steps;
    kernelCopy<<<(nmax + 255) / 256, 256, 0, stream>>>(rd, pa, na, out, n_steps);
}